// HardQuadLoss_1425929142961
// MI455X (gfx1250) — compile-verified
//
#include <hip/hip_runtime.h>

typedef __attribute__((ext_vector_type(2))) float v2f;
typedef __attribute__((ext_vector_type(8))) float v8f;

#define NROWS 8192
#define DIMS  256
#define STRIP 32           // rows per block (= rows per wave now)
#define LDS_STRIDE 260     // padded row stride in floats (260 % 64 == 4 -> conflict-free b64)
#define NCHUNK 8           // column chunks (grid.y)

// ---------------------------------------------------------------------------
// Kernel 1: row squared norms  +  init global reduction buffers
// ---------------------------------------------------------------------------
__global__ __launch_bounds__(256) void sq_init_kernel(
    const float* __restrict__ X, float* __restrict__ sq,
    unsigned* __restrict__ apBits, unsigned long long* __restrict__ anPack)
{
    const int wave = threadIdx.x >> 5;
    const int lane = threadIdx.x & 31;
    const int row  = blockIdx.x * 8 + wave;
    const float* p = X + (size_t)row * DIMS;
    float s = 0.f;
    #pragma unroll
    for (int k = 0; k < DIMS / 32; ++k) {
        float x = p[lane + 32 * k];
        s = fmaf(x, x, s);
    }
    #pragma unroll
    for (int m = 16; m >= 1; m >>= 1) s += __shfl_xor(s, m, 32);
    if (lane == 0) sq[row] = s;

    int gid = blockIdx.x * 256 + threadIdx.x;
    if (gid < NROWS) { apBits[gid] = 0u; anPack[gid] = ~0ull; }
}

// ---------------------------------------------------------------------------
// Kernel 2: streamed Gram via V_WMMA_F32_16X16X4_F32 + hardest pos/neg.
// Each wave: full 32-row strip (2 row tiles, 2 accumulators) x one column
// slice -> every B fragment feeds TWO WMMAs (halved gather traffic).
// ---------------------------------------------------------------------------
__global__ __launch_bounds__(256) void hard_dist_kernel(
    const float* __restrict__ X, const float* __restrict__ sq,
    const int* __restrict__ tgt,
    unsigned* __restrict__ apBits, unsigned long long* __restrict__ anPack)
{
    __shared__ float ldsA[STRIP * LDS_STRIDE];

    const int i0 = blockIdx.x * STRIP;

    // Stage the 32-row strip into padded LDS (float4 copies, coalesced).
    for (int t = threadIdx.x; t < STRIP * (DIMS / 4); t += 256) {
        int row = t >> 6;          // DIMS/4 == 64 float4 per row
        int c4  = t & 63;
        float4 vv = *(const float4*)(X + (size_t)(i0 + row) * DIMS + c4 * 4);
        *(float4*)(&ldsA[row * LDS_STRIDE + c4 * 4]) = vv;
    }
    __syncthreads();

    const int c    = threadIdx.x >> 5;   // 0..7 : column-tile interleave
    const int lane = threadIdx.x & 31;
    const int lm   = lane & 15;
    const int lh   = lane >> 4;

    // Per-lane constants for the 16 C-matrix rows this lane owns
    // (row tile rt in {0,1}, M = v + 8*lh).
    float sqrow[16]; int trow[16];
    #pragma unroll
    for (int rt = 0; rt < 2; ++rt)
        #pragma unroll
        for (int v = 0; v < 8; ++v) {
            int row = i0 + 16 * rt + 8 * lh + v;
            sqrow[rt * 8 + v] = sq[row];
            trow[rt * 8 + v]  = tgt[row];
        }

    float run_max[16], run_min[16]; int run_idx[16];
    #pragma unroll
    for (int u = 0; u < 16; ++u) {
        run_max[u] = 0.f;
        run_min[u] = __builtin_inff();
        run_idx[u] = 0;
    }

    // A fragment gather bases: lane -> row (tileBase + lm), K offset 2*lh.
    const float* aBase0 = &ldsA[lm * LDS_STRIDE + 2 * lh];
    const float* aBase1 = aBase0 + 16 * LDS_STRIDE;

    const int colsPerChunk  = NROWS / NCHUNK;     // 1024
    const int tilesPerChunk = colsPerChunk / 16;  // 64
    const int chunkBase     = blockIdx.y * colsPerChunk;

    for (int t = c; t < tilesPerChunk; t += 8) {
        const int j0   = chunkBase + t * 16;
        const int jcol = j0 + lm;                 // this lane's column (N = lm)
        const float sqc = sq[jcol];
        const int   tc  = tgt[jcol];
        // B fragment gather: identical per-lane pattern as A (Gram symmetry).
        const float* bBase = X + (size_t)jcol * DIMS + 2 * lh;

        v8f acc0 = {};
        v8f acc1 = {};
        #pragma unroll
        for (int kk = 0; kk < DIMS; kk += 4) {
            v2f b  = *(const v2f*)(bBase + kk);   // global_load_b64, L2-hot
            v2f a0 = *(const v2f*)(aBase0 + kk);  // ds_load_b64 (hoistable)
            v2f a1 = *(const v2f*)(aBase1 + kk);
            acc0 = __builtin_amdgcn_wmma_f32_16x16x4_f32(
                false, a0, false, b, (short)0, acc0, false, false);
            acc1 = __builtin_amdgcn_wmma_f32_16x16x4_f32(
                false, a1, false, b, (short)0, acc1, false, false);
        }

        #pragma unroll
        for (int rt = 0; rt < 2; ++rt)
            #pragma unroll
            for (int v = 0; v < 8; ++v) {
                float g    = (rt == 0) ? acc0[v] : acc1[v];
                int   u    = rt * 8 + v;
                float d2   = sqrow[u] + sqc - 2.0f * g;
                float dist = sqrtf(fmaxf(d2, 1e-12f));
                if (trow[u] == tc) {
                    run_max[u] = fmaxf(run_max[u], dist);
                } else if (dist < run_min[u]) {   // ascending j per lane -> first occurrence
                    run_min[u] = dist;
                    run_idx[u] = jcol;
                }
            }
    }

    // Reduce across the 16 lanes of each half-wave, then combine globally.
    #pragma unroll
    for (int rt = 0; rt < 2; ++rt)
        #pragma unroll
        for (int v = 0; v < 8; ++v) {
            int   u  = rt * 8 + v;
            float mx = run_max[u], mn = run_min[u];
            int   ix = run_idx[u];
            #pragma unroll
            for (int s = 1; s < 16; s <<= 1) {
                float omx = __shfl_xor(mx, s, 16);
                float omn = __shfl_xor(mn, s, 16);
                int   oix = __shfl_xor(ix, s, 16);
                mx = fmaxf(mx, omx);
                if (omn < mn || (omn == mn && oix < ix)) { mn = omn; ix = oix; }
            }
            if (lm == 0) {
                int row = i0 + 16 * rt + 8 * lh + v;
                atomicMax(&apBits[row], __float_as_uint(mx));
                unsigned long long key =
                    ((unsigned long long)__float_as_uint(mn) << 32) | (unsigned)ix;
                atomicMin(&anPack[row], key);
            }
        }
}

// ---------------------------------------------------------------------------
// Kernel 3: decode, gather dist_an[min_idx], final margin losses
// ---------------------------------------------------------------------------
__global__ __launch_bounds__(256) void loss_kernel(
    const unsigned* __restrict__ apBits,
    const unsigned long long* __restrict__ anPack,
    float* __restrict__ out)
{
    __shared__ float red[256];
    float s = 0.f;
    for (int i = threadIdx.x; i < NROWS; i += 256) {
        float ap = __uint_as_float(apBits[i]);
        unsigned long long k = anPack[i];
        float an  = __uint_as_float((unsigned)(k >> 32));
        int   idx = (int)(unsigned)(k & 0xFFFFFFFFull);
        unsigned long long k2 = anPack[idx];
        float an2 = __uint_as_float((unsigned)(k2 >> 32));
        s += fmaxf(ap - an  + 1.2f, 0.f);
        s += fmaxf(ap - an2 + 0.3f, 0.f);
    }
    red[threadIdx.x] = s;
    __syncthreads();
    #pragma unroll
    for (int m = 128; m >= 1; m >>= 1) {
        if (threadIdx.x < m) red[threadIdx.x] += red[threadIdx.x + m];
        __syncthreads();
    }
    if (threadIdx.x == 0) out[0] = red[0] / (float)NROWS;
}

// ---------------------------------------------------------------------------
extern "C" void kernel_launch(void* const* d_in, const int* in_sizes, int n_in,
                              void* d_out, int out_size, void* d_ws, size_t ws_size,
                              hipStream_t stream)
{
    const float* X   = (const float*)d_in[0];
    const int*   tgt = (const int*)d_in[1];
    float*       out = (float*)d_out;

    char* ws = (char*)d_ws;
    float*              sq     = (float*)ws;                       // 32 KB
    unsigned*           apBits = (unsigned*)(ws + 32768);          // 32 KB
    unsigned long long* anPack = (unsigned long long*)(ws + 65536);// 64 KB

    sq_init_kernel<<<NROWS / 8, 256, 0, stream>>>(X, sq, apBits, anPack);

    dim3 grid(NROWS / STRIP, NCHUNK);   // 256 x 8 blocks, 8 waves each
    hard_dist_kernel<<<grid, 256, 0, stream>>>(X, sq, tgt, apBits, anPack);

    loss_kernel<<<1, 256, 0, stream>>>(apBits, anPack, out);
}